// HeteroGraphEmbedding_72559177498820
// MI455X (gfx1250) — compile-verified
//
#include <hip/hip_runtime.h>

// Problem constants from the reference
#define N_SEG 8192
#define D_IN  64
#define OUT_F 128

typedef __attribute__((ext_vector_type(2))) float v2f;
typedef __attribute__((ext_vector_type(4))) float v4f;
typedef __attribute__((ext_vector_type(8))) float v8f;

#define NEG_BIG (-3.4028234e38f)

static __device__ __forceinline__ v4f v4max(v4f a, v4f b) {
    v4f r;
    r.x = fmaxf(a.x, b.x);
    r.y = fmaxf(a.y, b.y);
    r.z = fmaxf(a.z, b.z);
    r.w = fmaxf(a.w, b.w);
    return r;
}

// ---------------------------------------------------------------------------
// Phase 0: zero the 128-float output so atomicMax(int-bits) works (all values
// we feed it are clamped >= 0, so IEEE bits are monotone under int compare).
// ---------------------------------------------------------------------------
__global__ void hg_zero_out(float* __restrict__ out) {
    out[threadIdx.x] = 0.0f;
}

// ---------------------------------------------------------------------------
// Phase 1: streaming segment-max (the bandwidth-bound phase: ~1.02 GB read,
// ~44 us floor at 23.3 TB/s).  grid = (N_SEG, 2), block = 256.
//  Thread layout: col4 = (tid&15)*4  -> owns 4 consecutive columns (float4)
//                 rowlane = tid>>4   -> 16 row-lanes strided through segment
//  Rows are 256B-aligned so v4f NT loads are fully coalesced (16 lanes cover
//  one row).  Two independent accumulators keep >=2 NT loads in flight per
//  lane so the s_wait/v_max chain isn't on the critical path every 256B.
// ---------------------------------------------------------------------------
__global__ void hg_segmax(const float* __restrict__ xA,
                          const float* __restrict__ xB,
                          const int*   __restrict__ ptrA,
                          const int*   __restrict__ ptrB,
                          float*       __restrict__ m)  // [2*N_SEG][64]
{
    __shared__ v4f red[256];

    const int tid     = threadIdx.x;
    const int col4    = (tid & 15) * 4;
    const int rowlane = tid >> 4;
    const int seg     = blockIdx.x;
    const int type    = blockIdx.y;

    const float* __restrict__ x   = type ? xB   : xA;
    const int*   __restrict__ ptr = type ? ptrB : ptrA;

    const int start = ptr[seg];
    const int end   = ptr[seg + 1];

    v4f p0, p1;
    p0.x = p0.y = p0.z = p0.w = NEG_BIG;
    p1 = p0;

    int r = start + rowlane;
    for (; r + 16 < end; r += 32) {
        v4f v0 = __builtin_nontemporal_load((const v4f*)(x + (size_t)r        * D_IN + col4));
        v4f v1 = __builtin_nontemporal_load((const v4f*)(x + (size_t)(r + 16) * D_IN + col4));
        p0 = v4max(p0, v0);
        p1 = v4max(p1, v1);
    }
    if (r < end) {
        v4f v0 = __builtin_nontemporal_load((const v4f*)(x + (size_t)r * D_IN + col4));
        p0 = v4max(p0, v0);
    }
    v4f p = v4max(p0, p1);

    red[tid] = p;
    __syncthreads();
    // reduce across the 16 row-lanes (flat index = rowlane*16 + col16)
    for (int s = 128; s >= 16; s >>= 1) {
        if (tid < s) {
            p = v4max(p, red[tid + s]);
            red[tid] = p;
        }
        __syncthreads();
    }

    if (tid < 16) {
        v4f* dst = (v4f*)(m + ((size_t)type * N_SEG + seg) * D_IN + tid * 4);
        *dst = p;
    }
}

// ---------------------------------------------------------------------------
// Phase 2: per-type GEMM (m @ W^T) via V_WMMA_F32_16X16X4_F32, fused with
// bias + global column max + ReLU.
//
// One wave = one 16-wide output-column tile (o0..o0+15); its B tile (16 v2f
// fragments of W^T) lives in registers for the whole kernel.  It loops over 8
// consecutive 16-row tiles of M; per tile all 16 A fragments are loaded into
// a register array FIRST (one load clause, staggered waits) and then the 16
// chained k=4 WMMAs run, so the A fetch overlaps the previous tile's math.
//
// Fragment layouts (f32 16x16x4):
//  A: lane%16 = M row; lanes 0-15 carry K {kk,kk+1}, lanes 16-31 K {kk+2,kk+3}
//  B: lane%16 = N col; same K split (B[k][n] = W[o0+n][k])
//  D: lane%16 = column; 8 regs = rows 0-7 (lanes 0-15) / 8-15 (lanes 16-31)
//     -> per-lane max over 8 regs, then atomicMax across lane halves / tiles.
//
// 2 types * 8 col-tiles * 64 row-chunks = 1024 waves, 128 wmma each; M (4 MB)
// is L2-resident, so this phase is negligible next to phase 1.
// ---------------------------------------------------------------------------
__global__ void hg_wmma_linear_max(const float* __restrict__ m,   // [2*N_SEG][64]
                                   const float* __restrict__ Wa,  // [128][64]
                                   const float* __restrict__ Wb,
                                   const float* __restrict__ ba,  // [128]
                                   const float* __restrict__ bb,
                                   float*       __restrict__ out) // [128]
{
    const int tid   = threadIdx.x;
    const int lane  = tid & 31;
    const int l16   = lane & 15;
    const int half2 = (lane >> 4) * 2;      // K sub-offset: 0 or 2

    const int gw       = blockIdx.x * (blockDim.x >> 5) + (tid >> 5); // 0..1023
    const int type     = gw >> 9;           // 512 waves per type
    const int rem      = gw & 511;
    const int colTile  = rem >> 6;          // 0..7
    const int rowChunk = rem & 63;          // 0..63  (8 row-tiles each)

    const float* __restrict__ M    = m + (size_t)type * N_SEG * D_IN;
    const float* __restrict__ W    = type ? Wb : Wa;
    const float* __restrict__ bias = type ? bb : ba;

    const int    o0   = colTile * 16;
    const float* Wrow = W + (size_t)(o0 + l16) * D_IN;   // B column for this lane

    // Persistent B tile in registers (16 x v2f = 32 VGPRs)
    v2f bfrag[16];
    #pragma unroll
    for (int t = 0; t < 16; ++t)
        bfrag[t] = *(const v2f*)(Wrow + t * 4 + half2);

    float runmax = NEG_BIG;

    for (int rt = 0; rt < 8; ++rt) {
        const int    row  = (rowChunk * 8 + rt) * 16 + l16;   // A row for this lane
        const float* Mrow = M + (size_t)row * D_IN;

        // Fetch the full A tile first: one clause of 16 b64 loads.
        v2f afrag[16];
        #pragma unroll
        for (int t = 0; t < 16; ++t)
            afrag[t] = *(const v2f*)(Mrow + t * 4 + half2);

        v8f d = {0.f, 0.f, 0.f, 0.f, 0.f, 0.f, 0.f, 0.f};
        #pragma unroll
        for (int t = 0; t < 16; ++t) {
            d = __builtin_amdgcn_wmma_f32_16x16x4_f32(
                    /*neg_a=*/false, afrag[t], /*neg_b=*/false, bfrag[t],
                    /*c_mod=*/(short)0, d, /*reuse_a=*/false, /*reuse_b=*/false);
        }

        float t0 = fmaxf(fmaxf(fmaxf(d[0], d[1]), fmaxf(d[2], d[3])),
                         fmaxf(fmaxf(d[4], d[5]), fmaxf(d[6], d[7])));
        runmax = fmaxf(runmax, t0);
    }

    // bias is constant per column -> add after the row max; fold ReLU into the
    // max-with-zero; int-bits atomicMax is exact for non-negative floats.
    float val = fmaxf(runmax + bias[o0 + l16], 0.0f);
    atomicMax((int*)out + (o0 + l16), __float_as_int(val));
}

// ---------------------------------------------------------------------------
// Launcher.  Input order: x_a, x_b, W_a, W_b, b_a, b_b, ptr_a, ptr_b.
// Workspace usage: 2 * 8192 * 64 * 4 B = 4 MB for segment maxima.
// ---------------------------------------------------------------------------
extern "C" void kernel_launch(void* const* d_in, const int* in_sizes, int n_in,
                              void* d_out, int out_size, void* d_ws, size_t ws_size,
                              hipStream_t stream) {
    const float* xA   = (const float*)d_in[0];
    const float* xB   = (const float*)d_in[1];
    const float* Wa   = (const float*)d_in[2];
    const float* Wb   = (const float*)d_in[3];
    const float* ba   = (const float*)d_in[4];
    const float* bb   = (const float*)d_in[5];
    const int*   ptrA = (const int*)d_in[6];
    const int*   ptrB = (const int*)d_in[7];

    float* out = (float*)d_out;
    float* mws = (float*)d_ws;   // [2][N_SEG][64]

    hg_zero_out<<<1, OUT_F, 0, stream>>>(out);

    dim3 g1(N_SEG, 2, 1);
    hg_segmax<<<g1, 256, 0, stream>>>(xA, xB, ptrA, ptrB, mws);

    // 1024 waves total, 8 waves per 256-thread block -> 128 blocks
    hg_wmma_linear_max<<<128, 256, 0, stream>>>(mws, Wa, Wb, ba, bb, out);
}